// UltraMemoryEfficientBlock_12068858102134
// MI455X (gfx1250) — compile-verified
//
#include <hip/hip_runtime.h>
#include <math.h>

#define D_MODEL 2048
#define SEQ     4096
#define BATCH   4
#define RANK    8
#define RF      32
#define NTOK    (BATCH*SEQ)   /* 16384 tokens */
#define TILE    16
#define XS      2052          /* padded LDS row stride (floats) */
#define EPSV    1e-5f

typedef float v2f __attribute__((ext_vector_type(2)));
typedef float v4f __attribute__((ext_vector_type(4)));
typedef float v8f __attribute__((ext_vector_type(8)));
typedef int   v4i __attribute__((ext_vector_type(4)));

#define HAVE_ASYNC_LDS 0
#if defined(__has_builtin)
#if __has_builtin(__builtin_amdgcn_global_load_async_to_lds_b128)
#undef HAVE_ASYNC_LDS
#define HAVE_ASYNC_LDS 1
#endif
#endif

typedef __attribute__((address_space(1))) v4i gv4i;   // global int4
typedef __attribute__((address_space(3))) v4i lv4i;   // LDS int4

__device__ __forceinline__ gv4i* as_global_v4i(const void* p) {
  return (gv4i*)(unsigned long long)p;
}
__device__ __forceinline__ lv4i* as_lds_v4i(const void* p) {
  // low 32 bits of a generic LDS pointer are the LDS byte offset
  return (lv4i*)(unsigned int)(unsigned long long)p;
}

__device__ __forceinline__ void wait_async0() {
#if defined(__has_builtin) && __has_builtin(__builtin_amdgcn_s_wait_asynccnt)
  __builtin_amdgcn_s_wait_asynccnt(0);
#else
  asm volatile("s_wait_asynccnt 0" ::: "memory");
#endif
}

// ---------------- kernel 0: fold norm affine into W_in column sums ----------
__global__ __launch_bounds__(256) void k0_weight_sums(
    const float* __restrict__ gamma, const float* __restrict__ beta,
    const float* __restrict__ W_in, float* __restrict__ cw) {
  __shared__ float red[16][256];
  int tid = threadIdx.x;
  float ag[8], ab[8];
  #pragma unroll
  for (int r = 0; r < 8; ++r) { ag[r] = 0.f; ab[r] = 0.f; }
  for (int col = tid; col < D_MODEL; col += 256) {
    float g = gamma[col], be = beta[col];
    #pragma unroll
    for (int r = 0; r < 8; ++r) {
      float w = W_in[col * 8 + r];
      ag[r] += g * w; ab[r] += be * w;
    }
  }
  #pragma unroll
  for (int r = 0; r < 8; ++r) { red[r][tid] = ag[r]; red[8 + r][tid] = ab[r]; }
  __syncthreads();
  for (int s = 128; s > 0; s >>= 1) {
    if (tid < s)
      for (int q = 0; q < 16; ++q) red[q][tid] += red[q][tid + s];
    __syncthreads();
  }
  if (tid < 16) cw[tid] = red[tid][0];
}

// ---------------- kernel 0b: Wp = gamma (x) W_down (fold LN scale into B) ---
__global__ __launch_bounds__(256) void k0b_wdown_fold(
    const float* __restrict__ gamma, const float* __restrict__ W_down,
    float* __restrict__ wp) {
  int lin = (blockIdx.x * 256 + threadIdx.x) * 4;   // 2048*32 elements
  int k = lin >> 5;
  v4f w = *(const v4f*)(W_down + lin);
  float g = gamma[k];
  v4f o = {w[0] * g, w[1] * g, w[2] * g, w[3] * g};
  *(v4f*)(wp + lin) = o;
}

// ---------------- kernel 0c: per-column sums of gamma*W_down, beta*W_down ---
__global__ __launch_bounds__(256) void k0c_wdown_sums(
    const float* __restrict__ gamma, const float* __restrict__ beta,
    const float* __restrict__ W_down,
    float* __restrict__ cg2, float* __restrict__ cb2) {
  __shared__ float rg[256], rb[256];
  int tid = threadIdx.x;
  int n = tid & 31, kc = tid >> 5;
  float sg = 0.f, sb = 0.f;
  for (int k = kc * 256; k < kc * 256 + 256; ++k) {
    float w = W_down[k * 32 + n];
    sg += gamma[k] * w; sb += beta[k] * w;
  }
  rg[tid] = sg; rb[tid] = sb;
  __syncthreads();
  if (tid < 32) {
    float a = 0.f, b = 0.f;
    for (int j = 0; j < 8; ++j) { a += rg[tid + j * 32]; b += rb[tid + j * 32]; }
    cg2[tid] = a; cb2[tid] = b;
  }
}

// ---------------- kernel 1: single pass -> per-token stats + u = ln(x)@W_in -
__global__ __launch_bounds__(256) void k1_norm_proj(
    const float* __restrict__ x, const float* __restrict__ gamma,
    const float* __restrict__ W_in, const float* __restrict__ b_in,
    const float* __restrict__ cw,
    float* __restrict__ u_ws, float* __restrict__ st_ws) {
  __shared__ float red[10][256];
  int tid = threadIdx.x;
  long token = blockIdx.x;
  const float* xr = x + token * (long)D_MODEL;
  float s1 = 0.f, s2 = 0.f, aw[8];
  #pragma unroll
  for (int r = 0; r < 8; ++r) aw[r] = 0.f;
  #pragma unroll
  for (int j = 0; j < 2; ++j) {
    int col = (j * 256 + tid) * 4;
    v4f xv = *(const v4f*)(xr + col);
    v4f g  = *(const v4f*)(gamma + col);
    #pragma unroll
    for (int c2 = 0; c2 < 4; ++c2) {
      float xvc = xv[c2];
      s1 += xvc; s2 += xvc * xvc;
      float gx = xvc * g[c2];
      const v4f* wr = (const v4f*)(W_in + (long)(col + c2) * 8);
      v4f w0 = wr[0], w1 = wr[1];
      #pragma unroll
      for (int r = 0; r < 4; ++r) { aw[r] += gx * w0[r]; aw[4 + r] += gx * w1[r]; }
    }
  }
  red[0][tid] = s1; red[1][tid] = s2;
  #pragma unroll
  for (int r = 0; r < 8; ++r) red[2 + r][tid] = aw[r];
  __syncthreads();
  for (int s = 128; s > 0; s >>= 1) {
    if (tid < s)
      for (int q = 0; q < 10; ++q) red[q][tid] += red[q][tid + s];
    __syncthreads();
  }
  float mu  = red[0][0] * (1.f / D_MODEL);
  float var = red[1][0] * (1.f / D_MODEL) - mu * mu;
  float rs  = rsqrtf(var + EPSV);
  if (tid == 0) { st_ws[token * 2] = mu; st_ws[token * 2 + 1] = rs; }
  if (tid < 8)
    u_ws[token * 8 + tid] = rs * (red[2 + tid][0] - mu * cw[tid]) + cw[8 + tid] + b_in[tid];
}

// ---------------- kernel 2: semiseparable scan (chunked parallel) -----------
__global__ __launch_bounds__(256) void k2_scan(
    const float* __restrict__ decay_logit,
    const float* __restrict__ u_ws, float* __restrict__ h_ws) {
  __shared__ float ce[256], cin[256];
  int tid = threadIdx.x;
  int r = tid & 7, c = tid >> 3;
  long base = (long)blockIdx.x * SEQ * RANK;
  float dec = 1.f / (1.f + __expf(-decay_logit[r]));
  float d128 = dec;
  #pragma unroll
  for (int i = 0; i < 7; ++i) d128 *= d128;
  long cbase = base + (long)c * 128 * RANK + r;
  float hl = 0.f;
  for (int i = 0; i < 128; ++i) hl = dec * hl + u_ws[cbase + (long)i * RANK];
  ce[c * 8 + r] = hl;
  __syncthreads();
  if (tid < 8) {
    float H = 0.f;
    for (int cc = 0; cc < 32; ++cc) {
      cin[cc * 8 + tid] = H;
      H = d128 * H + ce[cc * 8 + tid];
    }
  }
  __syncthreads();
  float h = cin[c * 8 + r];
  for (int i = 0; i < 128; ++i) {
    long idx = cbase + (long)i * RANK;
    h = dec * h + u_ws[idx];
    h_ws[idx] = h;
  }
}

// ---------------- kernel 3: fused AR-out + LN2 + WMMA FFN + residual --------
__global__ __launch_bounds__(256) void k3_fused(
    const float* __restrict__ x,
    const float* __restrict__ gamma, const float* __restrict__ beta,
    const float* __restrict__ W_out, const float* __restrict__ b_out,
    const float* __restrict__ D_skip,
    const float* __restrict__ wp,           // gamma (x) W_down
    const float* __restrict__ cg2w, const float* __restrict__ cb2w,
    const float* __restrict__ b_down,
    const float* __restrict__ W_up, const float* __restrict__ b_up,
    const float* __restrict__ h_ws, const float* __restrict__ st_ws,
    float* __restrict__ out) {
  extern __shared__ float sm[];
  float* xnew = sm;                        // TILE*XS
  float* hrow = sm + TILE * XS;            // 128
  float* st1  = hrow + TILE * RANK;        // 32
  float* st2  = st1 + 32;                  // 32
  float* hf   = st2 + 32;                  // 16*33
  float* red  = hf + TILE * 33;            // 2048
  int tid = threadIdx.x;
  long token0 = (long)blockIdx.x * TILE;
  const float* xb = x + token0 * D_MODEL;
  float* ob = out + token0 * D_MODEL;

#if HAVE_ASYNC_LDS
  // ---- async DMA: stage contiguous 128KB x tile straight into LDS ----------
  #pragma unroll 4
  for (int it = 0; it < 32; ++it) {
    int lin = (it * 256 + tid) * 4;
    int r0 = lin >> 11, c0 = lin & 2047;
    __builtin_amdgcn_global_load_async_to_lds_b128(
        as_global_v4i(xb + lin), as_lds_v4i(xnew + r0 * XS + c0), 0, 0);
  }
#endif
  if (tid < TILE * RANK) hrow[tid] = h_ws[token0 * RANK + tid];
  if (tid >= 128 && tid < 160) st1[tid - 128] = st_ws[token0 * 2 + (tid - 128)];
#if HAVE_ASYNC_LDS
  wait_async0();
#endif
  __syncthreads();

  // ---- phase B: x_new = x + h@W_out + b_out + xn*D_skip ; LN2 moments ------
  int row = tid >> 4, cg = tid & 15;
  float mu = st1[row * 2], rs = st1[row * 2 + 1];
  float hr[8];
  #pragma unroll
  for (int r = 0; r < 8; ++r) hr[r] = hrow[row * 8 + r];
  float s1 = 0.f, s2 = 0.f;
  for (int kc = 0; kc < 16; ++kc) {
    int col0 = kc * 128 + cg * 8;
    v4f xa, xc;
#if HAVE_ASYNC_LDS
    xa = *(const v4f*)(xnew + row * XS + col0);
    xc = *(const v4f*)(xnew + row * XS + col0 + 4);
#else
    const v4f* xp = (const v4f*)(xb + (long)row * D_MODEL + col0);
    xa = xp[0]; xc = xp[1];
    if (kc < 15) __builtin_prefetch(xb + (long)row * D_MODEL + col0 + 128, 0, 1);
#endif
    float xv[8] = {xa[0], xa[1], xa[2], xa[3], xc[0], xc[1], xc[2], xc[3]};
    float res[8];
    #pragma unroll
    for (int c = 0; c < 8; ++c) {
      int col = col0 + c;
      float xn = (xv[c] - mu) * rs * gamma[col] + beta[col];
      float acc = b_out[col] + xn * D_skip[col];
      #pragma unroll
      for (int r = 0; r < 8; ++r) acc += hr[r] * W_out[r * D_MODEL + col];
      float nv = xv[c] + acc;
      res[c] = nv;
      s1 += nv; s2 += nv * nv;
    }
    *(v4f*)(xnew + row * XS + col0)     = (v4f){res[0], res[1], res[2], res[3]};
    *(v4f*)(xnew + row * XS + col0 + 4) = (v4f){res[4], res[5], res[6], res[7]};
  }
  red[tid] = s1; red[256 + tid] = s2;
  __syncthreads();
  if (cg == 0) {
    float a = 0.f, b2 = 0.f;
    for (int j = 0; j < 16; ++j) { a += red[row * 16 + j]; b2 += red[256 + row * 16 + j]; }
    float m2 = a * (1.f / D_MODEL);
    float v2 = b2 * (1.f / D_MODEL) - m2 * m2;
    st2[row * 2] = m2; st2[row * 2 + 1] = rsqrtf(v2 + EPSV);
  }
  __syncthreads();

  // ---- phase D: S = x_new @ (gamma (x) W_down), WMMA f32 16x16x4.
  //      LN affine applied in epilogue: hf = relu(r2*(S - m2*cg2) + cb2 + b_down)
  int wave = tid >> 5, lane = tid & 31;
  int lr = lane & 15, hi = lane >> 4;
  {
    int ntile = wave >> 2, kq = wave & 3;
    int kbeg = kq * 512;
    int n = ntile * 16 + lr;
    v8f c = {0.f, 0.f, 0.f, 0.f, 0.f, 0.f, 0.f, 0.f};
    for (int k = kbeg; k < kbeg + 512; k += 4) {
      int ka = k + hi * 2;   // lanes 0-15: K=k,k+1 ; lanes 16-31: K=k+2,k+3
      v2f A  = *(const v2f*)(xnew + lr * XS + ka);
      v2f Bv = {wp[(long)ka * RF + n], wp[(long)(ka + 1) * RF + n]};
      c = __builtin_amdgcn_wmma_f32_16x16x4_f32(false, A, false, Bv, (short)0, c, false, false);
    }
    #pragma unroll
    for (int j = 0; j < 8; ++j) red[wave * 256 + j * 32 + lane] = c[j];
  }
  __syncthreads();
  for (int e = tid; e < 512; e += 256) {
    int nt = e >> 8, idx = e & 255;
    int j = idx >> 5, ln = idx & 31;
    float ss = 0.f;
    #pragma unroll
    for (int q = 0; q < 4; ++q) ss += red[(nt * 4 + q) * 256 + j * 32 + ln];
    int rrow = j + ((ln >> 4) << 3);
    int ccol = nt * 16 + (ln & 15);
    float m2 = st2[rrow * 2], r2 = st2[rrow * 2 + 1];
    float val = r2 * (ss - m2 * cg2w[ccol]) + cb2w[ccol] + b_down[ccol];
    hf[rrow * 33 + ccol] = val > 0.f ? val : 0.f;
  }
  __syncthreads();

  // ---- phase E: accumulate hf @ W_up + b_up into the LDS x_new tile --------
  for (int i = 0; i < 16; ++i) {
    int col0 = (wave + i * 8) * 16;
    int coln = col0 + lr;
    v8f c = {0.f, 0.f, 0.f, 0.f, 0.f, 0.f, 0.f, 0.f};
    #pragma unroll
    for (int k = 0; k < 32; k += 4) {
      int ka = k + hi * 2;
      v2f A  = {hf[lr * 33 + ka], hf[lr * 33 + ka + 1]};
      v2f Bv = {W_up[(long)ka * D_MODEL + coln], W_up[(long)(ka + 1) * D_MODEL + coln]};
      c = __builtin_amdgcn_wmma_f32_16x16x4_f32(false, A, false, Bv, (short)0, c, false, false);
    }
    float bu = b_up[coln];
    #pragma unroll
    for (int j = 0; j < 8; ++j) {
      int rrow = j + hi * 8;                 // D layout: M=j / M=j+8 per half-wave
      atomicAdd(&xnew[rrow * XS + coln], c[j] + bu);   // ds_add_f32, single writer
    }
  }
  __syncthreads();

  // ---- phase F: stream finished tile out, coalesced non-temporal b128 -----
  #pragma unroll 4
  for (int it = 0; it < 32; ++it) {
    int lin = (it * 256 + tid) * 4;
    int r0 = lin >> 11, c0 = lin & 2047;
    v4f v = *(const v4f*)(xnew + r0 * XS + c0);
    __builtin_nontemporal_store(v, (v4f*)(ob + lin));
  }
}

extern "C" void kernel_launch(void* const* d_in, const int* in_sizes, int n_in,
                              void* d_out, int out_size, void* d_ws, size_t ws_size,
                              hipStream_t stream) {
  const float* x      = (const float*)d_in[0];
  const float* gamma  = (const float*)d_in[1];
  const float* beta   = (const float*)d_in[2];
  const float* W_in   = (const float*)d_in[3];
  const float* b_in   = (const float*)d_in[4];
  const float* dlogit = (const float*)d_in[5];
  const float* W_out  = (const float*)d_in[6];
  const float* b_out  = (const float*)d_in[7];
  const float* D_skip = (const float*)d_in[8];
  const float* W_down = (const float*)d_in[9];
  const float* b_down = (const float*)d_in[10];
  const float* W_up   = (const float*)d_in[11];
  const float* b_up   = (const float*)d_in[12];
  float* out = (float*)d_out;
  float* ws  = (float*)d_ws;

  float* u_ws  = ws;                                 // NTOK*RANK
  float* h_ws  = u_ws + (long)NTOK * RANK;           // NTOK*RANK
  float* st_ws = h_ws + (long)NTOK * RANK;           // NTOK*2
  float* cw_ws = st_ws + 2L * NTOK;                  // 16
  float* wp_ws = cw_ws + 16;                         // 2048*32
  float* cg2_ws = wp_ws + (long)D_MODEL * RF;        // 32
  float* cb2_ws = cg2_ws + 32;                       // 32

  k0_weight_sums<<<1, 256, 0, stream>>>(gamma, beta, W_in, cw_ws);
  k0b_wdown_fold<<<(D_MODEL * RF) / 1024, 256, 0, stream>>>(gamma, W_down, wp_ws);
  k0c_wdown_sums<<<1, 256, 0, stream>>>(gamma, beta, W_down, cg2_ws, cb2_ws);
  k1_norm_proj<<<NTOK, 256, 0, stream>>>(x, gamma, W_in, b_in, cw_ws, u_ws, st_ws);
  k2_scan<<<BATCH, 256, 0, stream>>>(dlogit, u_ws, h_ws);

  int smemBytes = (TILE * XS + TILE * RANK + 32 + 32 + TILE * 33 + 2048) * (int)sizeof(float);
  (void)hipFuncSetAttribute((const void*)k3_fused,
                            hipFuncAttributeMaxDynamicSharedMemorySize, smemBytes);
  k3_fused<<<NTOK / TILE, 256, smemBytes, stream>>>(
      x, gamma, beta, W_out, b_out, D_skip,
      wp_ws, cg2_ws, cb2_ws, b_down, W_up, b_up, h_ws, st_ws, out);
}